// LSTM_22170621182258
// MI455X (gfx1250) — compile-verified
//
#include <hip/hip_runtime.h>
#include <math.h>

// Problem constants (from reference): B=32, S=512, D=H=1024, L=2
#define B_   32
#define S_   512
#define D_   1024
#define H_   1024
#define G_   4096     // 4*H gate width
#define M_   16384    // B*S flattened rows
#define NWG  32       // persistent workgroups for recurrence (32 h-cols each)

typedef float v2f __attribute__((ext_vector_type(2)));
typedef float v8f __attribute__((ext_vector_type(8)));

// D = A(16x4, f32) * B(4x16, f32) + C(16x16, f32)  — native fp32 WMMA on CDNA5
__device__ __forceinline__ v8f wmma_f32(v2f a, v2f b, v8f c) {
    return __builtin_amdgcn_wmma_f32_16x16x4_f32(
        /*neg_a=*/false, a, /*neg_b=*/false, b,
        /*c_mod=*/(short)0, c, /*reuse_a=*/false, /*reuse_b=*/false);
}

struct Bar { unsigned count; unsigned gen; };

__device__ __forceinline__ void grid_barrier(Bar* bar, unsigned& mygen) {
    __syncthreads();
    if (threadIdx.x == 0) {
        unsigned g = mygen;
        if (atomicAdd(&bar->count, 1u) == NWG - 1u) {
            atomicExch(&bar->count, 0u);
            __threadfence();
            atomicAdd(&bar->gen, 1u);
        } else {
            while (atomicAdd(&bar->gen, 0u) == g) { __builtin_amdgcn_s_sleep(1); }
        }
        mygen = g + 1u;
    }
    __syncthreads();
}

__global__ void bar_init_kernel(Bar* bar) { bar->count = 0u; bar->gen = 0u; }

// out[M_ x G_] = A[M_ x 1024] @ W[1024 x G_] + bias
// Grid: (G_/128, M_/64), block 256 (8 waves). Wave -> 32x32 macro-tile:
// 2 row-tiles x 2 col-tiles of 16x16, 4 independent accumulators. Both A and
// B operands feed two WMMAs each (8 FLOP/byte from cache instead of 4).
__global__ __launch_bounds__(256) void xproj_kernel(
    const float* __restrict__ A, const float* __restrict__ W,
    const float* __restrict__ bias, float* __restrict__ out) {
    const int lane = threadIdx.x & 31, half = lane >> 4, l16 = lane & 15;
    const int wave = threadIdx.x >> 5;
    const int row0 = blockIdx.y * 64 + (wave & 1) * 32;    // 2 waves cover 64 rows
    const int col0 = blockIdx.x * 128 + (wave >> 1) * 32;  // 4 waves cover 128 cols

    v8f acc00 = {}, acc01 = {}, acc10 = {}, acc11 = {};
    const size_t arow0 = (size_t)(row0 + l16) * D_;
    const size_t arow1 = (size_t)(row0 + 16 + l16) * D_;

    for (int kko = 0; kko < D_; kko += 64) {
        // Stage next K-chunk of the weight panel into WGP cache (global_prefetch_b8)
        if (kko + 64 < D_) {
            __builtin_prefetch(&W[(size_t)(kko + 64 + (lane >> 1)) * G_
                                  + col0 + (lane & 1) * 16], 0, 1);
        }
        #pragma unroll 4
        for (int kk = kko; kk < kko + 64; kk += 4) {
            const int ak = kk + half * 2;       // A 16x4: lanes 0-15 K={0,1}, 16-31 K={2,3}
            v2f a0; a0.x = A[arow0 + ak]; a0.y = A[arow0 + ak + 1];
            v2f a1; a1.x = A[arow1 + ak]; a1.y = A[arow1 + ak + 1];
            const float* Wp = W + (size_t)ak * G_ + col0 + l16;  // B 4x16: VGPR v -> K = 2*half + v
            v2f b0; b0.x = Wp[0];  b0.y = Wp[G_];
            v2f b1; b1.x = Wp[16]; b1.y = Wp[G_ + 16];
            acc00 = wmma_f32(a0, b0, acc00);
            acc01 = wmma_f32(a0, b1, acc01);
            acc10 = wmma_f32(a1, b0, acc10);
            acc11 = wmma_f32(a1, b1, acc11);
        }
    }
    const float bz0 = bias[col0 + l16];
    const float bz1 = bias[col0 + 16 + l16];
    #pragma unroll
    for (int v = 0; v < 8; ++v) {               // C: VGPR v -> M = v + 8*half, N = l16
        const int r0 = row0 + v + half * 8;
        const int r1 = r0 + 16;
        out[(size_t)r0 * G_ + col0 + l16]      = acc00[v] + bz0;
        out[(size_t)r0 * G_ + col0 + 16 + l16] = acc01[v] + bz1;
        out[(size_t)r1 * G_ + col0 + l16]      = acc10[v] + bz0;
        out[(size_t)r1 * G_ + col0 + 16 + l16] = acc11[v] + bz1;
    }
}

// Persistent recurrence kernel. NWG blocks x 512 threads (16 wave32).
// Block owns h-cols [hc0, hc0+32) -> gate cols {g*1024+hc0 .. +32} for g=0..3.
// Wave (mr, g, nh) computes one 16x16 gate tile over K=1024 each step.
// c-state lives in LDS for the whole kernel; h goes through global + grid barrier.
__global__ __launch_bounds__(512) void lstm_rec_kernel(
    const float* __restrict__ xproj,  // [M_ x G_], bias already folded in
    const float* __restrict__ Whh,    // [H_ x G_]
    const float* __restrict__ h0, const float* __restrict__ c0,   // [B_ x H_]
    float* __restrict__ h_glb,        // [B_ x H_] scratch
    float* __restrict__ y,            // [B_ x S_ x H_]
    float* __restrict__ hn, float* __restrict__ cn,               // [B_ x H_]
    Bar* bar) {
    __shared__ float g_lds[4][B_][33];   // activated gates, padded vs bank conflicts
    __shared__ float c_lds[B_][33];      // persistent cell state slice

    const int tid  = threadIdx.x;
    const int lane = tid & 31, half = lane >> 4, l16 = lane & 15;
    const int wave = tid >> 5;
    const int mr = wave & 1, gg = (wave >> 1) & 3, nh = (wave >> 3) & 1;
    const int hc0 = blockIdx.x * 32;
    unsigned mygen = 0u;

    for (int idx = tid; idx < B_ * 32; idx += 512) {
        const int row = idx >> 5, hc = idx & 31;
        c_lds[row][hc] = c0[row * H_ + hc0 + hc];
        h_glb[row * H_ + hc0 + hc] = h0[row * H_ + hc0 + hc];
    }
    __threadfence();
    grid_barrier(bar, mygen);

    const int colg = gg * H_ + hc0 + nh * 16 + l16;   // this wave's gate column
    const size_t arow = (size_t)(mr * 16 + l16) * H_;

    for (int t = 0; t < S_; ++t) {
        // gates_tile = h @ Whh[:, colg-tile]  (two interleaved accumulators)
        v8f acc0 = {}, acc1 = {};
        #pragma unroll 2
        for (int kk = 0; kk < H_; kk += 8) {
            const int ak0 = kk + half * 2;
            const int ak1 = kk + 4 + half * 2;
            v2f a0; a0.x = h_glb[arow + ak0]; a0.y = h_glb[arow + ak0 + 1];
            v2f a1; a1.x = h_glb[arow + ak1]; a1.y = h_glb[arow + ak1 + 1];
            const float* Wp0 = Whh + (size_t)ak0 * G_ + colg;
            const float* Wp1 = Whh + (size_t)ak1 * G_ + colg;
            v2f b0v; b0v.x = Wp0[0]; b0v.y = Wp0[G_];
            v2f b1v; b1v.x = Wp1[0]; b1v.y = Wp1[G_];
            acc0 = wmma_f32(a0, b0v, acc0);
            acc1 = wmma_f32(a1, b1v, acc1);
        }
        // add precomputed input projection, apply activation, exchange via LDS
        #pragma unroll
        for (int v = 0; v < 8; ++v) {
            const int row = mr * 16 + v + half * 8;           // batch index
            float val = acc0[v] + acc1[v]
                      + xproj[((size_t)row * S_ + t) * G_ + colg];
            float act = (gg == 2) ? tanhf(val) : 1.f / (1.f + __expf(-val));
            g_lds[gg][row][nh * 16 + l16] = act;
        }
        __syncthreads();
        // pointwise cell/hidden update for this block's 32x32 (row, hcol) slice
        for (int idx = tid; idx < B_ * 32; idx += 512) {
            const int row = idx >> 5, hc = idx & 31;
            const float i = g_lds[0][row][hc], f = g_lds[1][row][hc];
            const float gv = g_lds[2][row][hc], o = g_lds[3][row][hc];
            const float cc = f * c_lds[row][hc] + i * gv;
            c_lds[row][hc] = cc;
            const float hh = o * tanhf(cc);
            h_glb[row * H_ + hc0 + hc] = hh;
            y[((size_t)row * S_ + t) * H_ + hc0 + hc] = hh;
        }
        __threadfence();
        grid_barrier(bar, mygen);   // full h visible device-wide before next step
    }

    for (int idx = tid; idx < B_ * 32; idx += 512) {
        const int row = idx >> 5, hc = idx & 31;
        hn[row * H_ + hc0 + hc] = h_glb[row * H_ + hc0 + hc];
        cn[row * H_ + hc0 + hc] = c_lds[row][hc];
    }
}

extern "C" void kernel_launch(void* const* d_in, const int* in_sizes, int n_in,
                              void* d_out, int out_size, void* d_ws, size_t ws_size,
                              hipStream_t stream) {
    (void)in_sizes; (void)n_in; (void)out_size; (void)ws_size;
    const float* x     = (const float*)d_in[0];
    const float* h0    = (const float*)d_in[1];   // [2,B,H]
    const float* c0    = (const float*)d_in[2];   // [2,B,H]
    const float* w_ih0 = (const float*)d_in[3];
    const float* w_hh0 = (const float*)d_in[4];
    const float* b0    = (const float*)d_in[5];
    const float* w_ih1 = (const float*)d_in[6];
    const float* w_hh1 = (const float*)d_in[7];
    const float* b1    = (const float*)d_in[8];

    float* out_y  = (float*)d_out;                 // [B,S,H]
    float* out_hn = out_y + (size_t)M_ * H_;       // [2,B,H]
    float* out_cn = out_hn + 2 * B_ * H_;          // [2,B,H]

    float* ws     = (float*)d_ws;
    float* xproj  = ws;                            // M_*G_ floats (256 MB)
    float* y0     = xproj + (size_t)M_ * G_;       // M_*H_ floats (64 MB)
    float* h_glb  = y0 + (size_t)M_ * H_;          // B_*H_ floats
    Bar*   bar    = (Bar*)(h_glb + B_ * H_);

    const dim3 ggrid(G_ / 128, M_ / 64);

    // ---- layer 0 ----
    bar_init_kernel<<<1, 1, 0, stream>>>(bar);
    xproj_kernel<<<ggrid, 256, 0, stream>>>(x, w_ih0, b0, xproj);
    lstm_rec_kernel<<<NWG, 512, 0, stream>>>(xproj, w_hh0, h0, c0,
                                             h_glb, y0, out_hn, out_cn, bar);
    // ---- layer 1 ----
    bar_init_kernel<<<1, 1, 0, stream>>>(bar);
    xproj_kernel<<<ggrid, 256, 0, stream>>>(y0, w_ih1, b1, xproj);
    lstm_rec_kernel<<<NWG, 512, 0, stream>>>(xproj, w_hh1, h0 + B_ * H_, c0 + B_ * H_,
                                             h_glb, out_y, out_hn + B_ * H_,
                                             out_cn + B_ * H_, bar);
}